// _YOLO_72722386256561
// MI455X (gfx1250) — compile-verified
//
#include <hip/hip_runtime.h>
#include <math.h>

// Problem constants (fixed by the reference).
#define BB 16
#define HH 64
#define WW 64
#define AA 5
#define CC 80
#define MM 32
#define HWC (HH * WW)

typedef __attribute__((ext_vector_type(2))) float v2f;
typedef __attribute__((ext_vector_type(8))) float v8f;

#if defined(__has_builtin)
#if __has_builtin(__builtin_amdgcn_wmma_f32_16x16x4_f32)
#define HAVE_WMMA_F32 1
#endif
#endif

// One block per image. Wave 0 does GT->anchor matching; all 8 waves share the
// 32x80 classification squared-error reduction (via f32 WMMA row-sum trick).
__global__ __launch_bounds__(256) void yolo_loss_kernel(
    const float* __restrict__ pred_boxes,   // (B,H,W,A,4) cx,cy,w,h
    const float* __restrict__ pred_o,       // (B,H,W,A)
    const float* __restrict__ pred_scores,  // (B,H,W,A,C)
    const float* __restrict__ gt_boxes,     // (B,M,4) xyxy
    const int*   __restrict__ gt_labels,    // (B,M)
    float* __restrict__ partials)           // (B,4) obj,noobj,bbox,clf
{
  __shared__ int   sh_matched[MM];
  __shared__ int   sh_label[MM];
  __shared__ float sh_obj[MM];
  __shared__ float sh_bbox[MM];
  __shared__ float sh_po2[MM];
  __shared__ float sh_noobj[MM];
  __shared__ float sh_lane[256];

  const int b   = blockIdx.x;
  const int tid = threadIdx.x;

  const float* pb_img = pred_boxes  + (size_t)b * HWC * AA * 4;
  const float* po_img = pred_o      + (size_t)b * HWC * AA;
  const float* ps_img = pred_scores + (size_t)b * HWC * AA * CC;

  // ---------- Phase 1: per-GT matching (wave 0, one lane per GT) ----------
  if (tid < MM) {
    const int m = tid;
    const float* gb = gt_boxes + ((size_t)b * MM + m) * 4;
    float gx1 = gb[0], gy1 = gb[1], gx2 = gb[2], gy2 = gb[3];
    float gcx = (gx1 + gx2) * 0.5f * (float)WW;
    float gcy = (gy1 + gy2) * 0.5f * (float)HH;
    float fcx = floorf(gcx), fcy = floorf(gcy);
    int   cell = (int)fcy * WW + (int)fcx;
    float area_g = (gx2 - gx1) * (gy2 - gy1);

    int   best_a  = 0;
    float best_iou = -1.0f;  // strict '>' -> first occurrence of max (jnp.argmax)
    for (int a = 0; a < AA; ++a) {
      const float* pb = pb_img + (size_t)(cell * AA + a) * 4;
      float pcx = pb[0], pcy = pb[1], pw = pb[2], ph = pb[3];
      float px1 = pcx - pw * 0.5f, py1 = pcy - ph * 0.5f;
      float px2 = pcx + pw * 0.5f, py2 = pcy + ph * 0.5f;
      float area_p = (px2 - px1) * (py2 - py1);
      float lx = fmaxf(px1, gx1), ly = fmaxf(py1, gy1);
      float rx = fminf(px2, gx2), ry = fminf(py2, gy2);
      float iw = fmaxf(rx - lx, 0.0f), ih = fmaxf(ry - ly, 0.0f);
      float inter = iw * ih;
      float iou = inter / (area_p + area_g - inter);
      if (iou > best_iou) { best_iou = iou; best_a = a; }
    }

    int   matched = cell * AA + best_a;
    float posel   = po_img[matched];
    float d_obj   = posel - best_iou;

    const float* sb = pb_img + (size_t)matched * 4;
    float scx = sb[0], scy = sb[1], sw = sb[2], shh = sb[3];
    float fx = scx * (float)WW; fx -= floorf(fx);
    float fy = scy * (float)HH; fy -= floorf(fy);
    float gfx = gcx - fcx;
    float gfy = gcy - fcy;
    float dx  = fx - gfx, dy = fy - gfy;
    float dwx = sqrtf(sw)  - sqrtf(gx2 - gx1);
    float dwy = sqrtf(shh) - sqrtf(gy2 - gy1);

    sh_matched[m] = matched;
    sh_label[m]   = gt_labels[(size_t)b * MM + m];
    sh_obj[m]     = d_obj * d_obj;
    sh_bbox[m]    = dx * dx + dy * dy + dwx * dwx + dwy * dwy;
    sh_po2[m]     = posel * posel;
  }
  __syncthreads();

  // noobj term: po^2 over UNIQUE matched positions (mirrors the scatter mask).
  if (tid < MM) {
    int mine = sh_matched[tid];
    bool unique = true;
    for (int mp = 0; mp < tid; ++mp)
      if (sh_matched[mp] == mine) { unique = false; break; }
    sh_noobj[tid] = unique ? sh_po2[tid] : 0.0f;
  }
  __syncthreads();

  // ---------- Phase 2: clf = sum((sel_scores - onehot)^2) over 32x80 ----------
  // WMMA f32 16x16x4 with B = ones: D[i,j] += rowsum(A[i,:]).  Every row-sum is
  // replicated over 16 columns, so (sum of all D lanes) / 16 = total reduction.
  const int wave  = tid >> 5;
  const int lane  = tid & 31;
  const int row   = lane & 15;   // A-matrix M index held by this lane
  const int khalf = lane >> 4;   // lanes 16-31 hold K=2,3 (ISA 16x4 A layout)
  float lane_sum = 0.0f;

#ifdef HAVE_WMMA_F32
  {
    v8f acc0 = {0, 0, 0, 0, 0, 0, 0, 0};  // GT rows 0..15
    v8f acc1 = {0, 0, 0, 0, 0, 0, 0, 0};  // GT rows 16..31
    v2f onesv; onesv.x = 1.0f; onesv.y = 1.0f;
    for (int t = wave; t < CC / 4; t += 8) {   // 20 K-chunks split over 8 waves
      int c = t * 4 + khalf * 2;
      {
        int m = row;
        int base = sh_matched[m] * CC;
        int lab  = sh_label[m];
        float s0 = ps_img[base + c]     - ((c     == lab) ? 1.0f : 0.0f);
        float s1 = ps_img[base + c + 1] - ((c + 1 == lab) ? 1.0f : 0.0f);
        v2f av; av.x = s0 * s0; av.y = s1 * s1;
        acc0 = __builtin_amdgcn_wmma_f32_16x16x4_f32(
            false, av, false, onesv, (short)0, acc0, false, false);
      }
      {
        int m = row + 16;
        int base = sh_matched[m] * CC;
        int lab  = sh_label[m];
        float s0 = ps_img[base + c]     - ((c     == lab) ? 1.0f : 0.0f);
        float s1 = ps_img[base + c + 1] - ((c + 1 == lab) ? 1.0f : 0.0f);
        v2f av; av.x = s0 * s0; av.y = s1 * s1;
        acc1 = __builtin_amdgcn_wmma_f32_16x16x4_f32(
            false, av, false, onesv, (short)0, acc1, false, false);
      }
    }
    for (int r = 0; r < 8; ++r) lane_sum += acc0[r] + acc1[r];
    lane_sum *= (1.0f / 16.0f);  // each row-sum appears in 16 columns
  }
#else
  // VALU fallback with identical work partitioning.
  for (int t = wave; t < CC / 4; t += 8) {
    int c = t * 4 + khalf * 2;
    for (int half = 0; half < 2; ++half) {
      int m = row + half * 16;
      int base = sh_matched[m] * CC;
      int lab  = sh_label[m];
      float s0 = ps_img[base + c]     - ((c     == lab) ? 1.0f : 0.0f);
      float s1 = ps_img[base + c + 1] - ((c + 1 == lab) ? 1.0f : 0.0f);
      lane_sum += s0 * s0 + s1 * s1;
    }
  }
#endif
  sh_lane[tid] = lane_sum;
  __syncthreads();

  // Deterministic in-order block reduction by thread 0 (tiny, fixed order).
  if (tid == 0) {
    float obj = 0.0f, noobj = 0.0f, bbox = 0.0f, clf = 0.0f;
    for (int m = 0; m < MM; ++m) {
      obj += sh_obj[m]; noobj += sh_noobj[m]; bbox += sh_bbox[m];
    }
    for (int i = 0; i < 256; ++i) clf += sh_lane[i];
    float* p = partials + b * 4;
    p[0] = obj; p[1] = noobj; p[2] = bbox; p[3] = clf;
  }
}

// Fixed-order cross-image reduction + loss scaling (deterministic, no atomics).
__global__ void yolo_finalize(const float* __restrict__ partials,
                              float* __restrict__ out)
{
  int j = threadIdx.x;
  if (j < 4) {
    float s = 0.0f;
    for (int b = 0; b < BB; ++b) s += partials[b * 4 + j];
    float scale = (j == 1) ? (0.5f / (float)BB)
                : (j == 2) ? (5.0f / (float)BB)
                           : (1.0f / (float)BB);
    out[j] = s * scale;
  }
}

extern "C" void kernel_launch(void* const* d_in, const int* in_sizes, int n_in,
                              void* d_out, int out_size, void* d_ws, size_t ws_size,
                              hipStream_t stream) {
  const float* pred_boxes  = (const float*)d_in[0];
  const float* pred_o      = (const float*)d_in[1];
  const float* pred_scores = (const float*)d_in[2];
  const float* gt_boxes    = (const float*)d_in[3];
  const int*   gt_labels   = (const int*)d_in[4];
  float* partials = (float*)d_ws;  // B*4 floats, fully overwritten each call

  yolo_loss_kernel<<<BB, 256, 0, stream>>>(pred_boxes, pred_o, pred_scores,
                                           gt_boxes, gt_labels, partials);
  yolo_finalize<<<1, 32, 0, stream>>>(partials, (float*)d_out);
}